// SpectralPredictor_34900904248012
// MI455X (gfx1250) — compile-verified
//
#include <hip/hip_runtime.h>

// ---------------- problem constants (match reference) ----------------
#define ZB   200
#define YB   192
#define XB   192
#define TT   (YB * XB)        // 36864 pixels per band
#define PB   15               // prediction bands
#define NWGT (PB + 4)         // 19 weights per band
#define LRc  0.01f
#define EPSc 1e-8f
#define MAXVc 32767.0f
#define MINVc (-32768.0f)

typedef float v2f __attribute__((ext_vector_type(2)));
typedef float v8f __attribute__((ext_vector_type(8)));

// =====================================================================
// Kernel 1: C[z,t] = sum_{k=1..14} weights0[z,3+k] * image[base(z)+k, t]
// (the non-adaptive part of the 19-dim dot). 16(band) x 16(pixel) x K=32
// fp32 GEMM tiles via V_WMMA_F32_16X16X4_F32, 8 K-chunks per tile,
// 4 pixel tiles per wave (A built once per wave).
// =====================================================================
__global__ __launch_bounds__(256)
void spectral_const_wmma(const float* __restrict__ img,
                         const float* __restrict__ wgt,
                         float* Cout) {
  const int lane = threadIdx.x & 31;
  const int wave = threadIdx.x >> 5;
  const int half = lane >> 4;          // 0: lanes 0-15, 1: lanes 16-31
  const int l15  = lane & 15;

  const int z0     = blockIdx.y * 16;
  const int kstart = (z0 - (PB - 1) > 0) ? (z0 - (PB - 1)) : 0;
  const bool tail  = (kstart + 31 >= ZB);          // only last band group

  // ---- A (16x4 f32 per chunk), fully branch-free ----
  const int zA    = z0 + l15;
  const int zc    = (zA < ZB) ? zA : (ZB - 1);
  const int baseZ = (zc - PB > 0) ? (zc - PB) : 0;
  const int mZ    = (zc < PB) ? zc : PB;
  v2f a[8];
#pragma unroll
  for (int c = 0; c < 8; ++c) {
#pragma unroll
    for (int e = 0; e < 2; ++e) {
      const int j   = 4 * c + 2 * half + e;
      const int k   = kstart + j - baseZ;
      const int kcl = (k < 1) ? 1 : ((k > PB - 1) ? (PB - 1) : k);
      const float cf = wgt[zc * NWGT + 3 + kcl];          // always in-bounds
      const bool ok  = (zA < ZB) && (k >= 1) && (k <= PB - 1) && (k < mZ);
      const float v  = ok ? cf : 0.0f;
      if (e == 0) a[c].x = v; else a[c].y = v;
    }
  }

  const int tileBase = (blockIdx.x * 8 + wave) * (16 * 4);  // 4 tiles/wave
  for (int tt = 0; tt < 4; ++tt) {
    const int nCol = tileBase + tt * 16 + l15;

    // ---- B (4x16 f32 per chunk): batch all 16 loads, then 8 WMMAs ----
    v2f b[8];
    if (!tail) {
      // one base address, all offsets are compile-time immediates
      const float* bb = img + (size_t)(kstart + 2 * half) * TT + nCol;
#pragma unroll
      for (int c = 0; c < 8; ++c) {
        b[c].x = bb[(size_t)c * 4 * TT];
        b[c].y = bb[(size_t)c * 4 * TT + TT];
      }
    } else {
      // clamped addresses + selects: branch-free, EXEC stays full
#pragma unroll
      for (int c = 0; c < 8; ++c) {
        const int s  = kstart + 4 * c + 2 * half;
        const int s0 = (s     < ZB) ? s       : (ZB - 1);
        const int s1 = (s + 1 < ZB) ? (s + 1) : (ZB - 1);
        const float bx = img[(size_t)s0 * TT + nCol];
        const float by = img[(size_t)s1 * TT + nCol];
        b[c].x = (s     < ZB) ? bx : 0.0f;
        b[c].y = (s + 1 < ZB) ? by : 0.0f;
      }
    }

    v8f acc = {0.f, 0.f, 0.f, 0.f, 0.f, 0.f, 0.f, 0.f};
#pragma unroll
    for (int c = 0; c < 8; ++c)
      acc = __builtin_amdgcn_wmma_f32_16x16x4_f32(
          false, a[c], false, b[c], (short)0, acc, false, false);

    // ---- D store: uniform fast path, one base + immediate offsets ----
    if (z0 + 16 <= ZB) {
      float* cb = Cout + (size_t)(z0 + 8 * half) * TT + nCol;
#pragma unroll
      for (int r = 0; r < 8; ++r)
        cb[(size_t)r * TT] = acc[r];
    } else {
#pragma unroll
      for (int r = 0; r < 8; ++r) {
        const int zs = z0 + r + half * 8;
        if (zs < ZB) Cout[(size_t)zs * TT + nCol] = acc[r];
      }
    }
  }
}

// =====================================================================
// Kernel 2: the serial adaptive scan (latency-bound; one lane per band).
// =====================================================================
__device__ __forceinline__ float clampf(float x, float lo, float hi) {
  return __builtin_amdgcn_fmed3f(x, lo, hi);     // single v_med3_f32
}

// One pixel step. u_i = (LR*d_i)*rcp(|d_i|+eps) keeps the w-chain at
// mul -> fma -> add -> med3 -> sub -> fma -> med3 (7 dependent VALU).
#define STEP(CE, NE, SE, TE, PX, RX, FX)                                        \
  {                                                                             \
    const float c_   = (CE);                                                    \
    const float Nv_  = (NE);                                                    \
    const float Sv_  = hasS ? (SE) : 0.0f;                                      \
    const float Ct_  = (TE);                                                    \
    const float Wl_  = (FX) ? 0.0f : Wv;                                        \
    const float NWl_ = (FX) ? 0.0f : NWv;                                       \
    const float d1_ = Nv_ - Wl_;                                                \
    const float d2_ = Wl_ - NWl_;                                               \
    const float d3_ = NWl_ - Nv_;                                               \
    const float d4_ = d2_ - d3_;          /* N + W - 2*NW */                    \
    const float u1_ = (LRc * d1_) * __builtin_amdgcn_rcpf(__builtin_fabsf(d1_) + EPSc); \
    const float u2_ = (LRc * d2_) * __builtin_amdgcn_rcpf(__builtin_fabsf(d2_) + EPSc); \
    const float u3_ = (LRc * d3_) * __builtin_amdgcn_rcpf(__builtin_fabsf(d3_) + EPSc); \
    const float u4_ = (LRc * d4_) * __builtin_amdgcn_rcpf(__builtin_fabsf(d4_) + EPSc); \
    const float pA_   = __builtin_fmaf(w1, Wl_, w0 * Nv_);                      \
    const float pB_   = __builtin_fmaf(w3, Sv_, __builtin_fmaf(w2, NWl_, Ct_)); \
    const float pred_ = clampf(pA_ + pB_, MINVc, MAXVc);                        \
    const float res_  = c_ - pred_;                                             \
    w0 = clampf(__builtin_fmaf(res_, u1_, w0), -1.0f, 1.0f);                    \
    w1 = clampf(__builtin_fmaf(res_, u2_, w1), -1.0f, 1.0f);                    \
    w2 = clampf(__builtin_fmaf(res_, u3_, w2), -1.0f, 1.0f);                    \
    w3 = clampf(__builtin_fmaf(res_, u4_, w3), -1.0f, 1.0f);                    \
    (PX) = pred_; (RX) = res_;                                                  \
    Wv = c_; NWv = Nv_;                                                         \
  }

template <bool FIRSTROW>
__device__ __forceinline__ void scan_row(
    const float* __restrict__ rc, const float* __restrict__ rn,
    const float* __restrict__ rs, const float* rC,  // rC may alias rr
    bool hasS, float& w0, float& w1, float& w2, float& w3,
    float* __restrict__ rp, float* rr) {
  float Wv = 0.f, NWv = 0.f;
  float4 t4 = *(const float4*)rC;                  // pipelined C stream
  for (int g = 0; g < XB / 4; ++g) {
    const float4 c4 = *(const float4*)(rc + 4 * g);
    float4 n4;
    if (FIRSTROW) n4 = make_float4(0.f, 0.f, 0.f, 0.f);
    else          n4 = *(const float4*)(rn + 4 * g);
    const float4 s4 = *(const float4*)(rs + 4 * g);
    const int   gn  = (g < XB / 4 - 1) ? (g + 1) : g;
    const float4 t4n = *(const float4*)(rC + 4 * gn);  // issued before stores
    float4 p4, r4;
    STEP(c4.x, n4.x, s4.x, t4.x, p4.x, r4.x, (g == 0));
    STEP(c4.y, n4.y, s4.y, t4.y, p4.y, r4.y, false);
    STEP(c4.z, n4.z, s4.z, t4.z, p4.z, r4.z, false);
    STEP(c4.w, n4.w, s4.w, t4.w, p4.w, r4.w, false);
    *(float4*)(rp + 4 * g) = p4;
    *(float4*)(rr + 4 * g) = r4;
    t4 = t4n;
  }
}

__global__ __launch_bounds__(32)
void spectral_scan(const float* __restrict__ img,
                   const float* __restrict__ wgt,
                   const float* Cpl,        // may alias resOut -> no restrict
                   float* predOut, float* resOut) {
  const int z = blockIdx.x * 32 + threadIdx.x;
  if (z >= ZB) return;

  float w0 = wgt[z * NWGT + 0];
  float w1 = wgt[z * NWGT + 1];
  float w2 = wgt[z * NWGT + 2];
  float w3 = wgt[z * NWGT + 3];

  const size_t zoff = (size_t)z * TT;
  const float* cur  = img + zoff;
  const float* sp   = img + (size_t)((z - PB > 0) ? (z - PB) : 0) * TT;
  const bool   hasS = (z >= 1);
  const float* Cc   = Cpl + zoff;
  float*       po   = predOut + zoff;
  float*       ro   = resOut + zoff;

  scan_row<true>(cur, cur, sp, Cc, hasS, w0, w1, w2, w3, po, ro);
  for (int y = 1; y < YB; ++y) {
    const int o = y * XB;
    if (y + 1 < YB) {                        // pull next row toward L0
      __builtin_prefetch(cur + o + XB, 0, 3);
      __builtin_prefetch(sp  + o + XB, 0, 3);
      __builtin_prefetch(Cc  + o + XB, 0, 3);
    }
    scan_row<false>(cur + o, cur + o - XB, sp + o, Cc + o, hasS,
                    w0, w1, w2, w3, po + o, ro + o);
  }
}

// =====================================================================
extern "C" void kernel_launch(void* const* d_in, const int* in_sizes, int n_in,
                              void* d_out, int out_size, void* d_ws, size_t ws_size,
                              hipStream_t stream) {
  (void)in_sizes; (void)n_in; (void)out_size;
  const float* image    = (const float*)d_in[0];
  const float* weights0 = (const float*)d_in[1];

  float* preds = (float*)d_out;
  float* ress  = (float*)d_out + (size_t)ZB * TT;

  // C(z,t) scratch (29.5 MB): prefer d_ws; else stage in the res half of
  // d_out (kernel 2 reads C[t] before overwriting that address with res[t],
  // same lane, program order -> safe).
  const size_t cbytes = (size_t)ZB * TT * sizeof(float);
  float* Cbuf = (ws_size >= cbytes) ? (float*)d_ws : ress;

  // Phase 1: WMMA fp32 GEMM. 13 band groups x 72 blocks x 8 waves x 4 tiles.
  dim3 g1(TT / (16 * 8 * 4), (ZB + 15) / 16);   // (72, 13)
  spectral_const_wmma<<<g1, 256, 0, stream>>>(image, weights0, Cbuf);

  // Phase 2: latency-bound serial scan, one lane per band, one wave/block.
  spectral_scan<<<(ZB + 31) / 32, 32, 0, stream>>>(image, weights0, Cbuf,
                                                   preds, ress);
}